// Memory_49417893707927
// MI455X (gfx1250) — compile-verified
//
#include <hip/hip_runtime.h>
#include <hip/hip_bf16.h>
#include <math.h>

typedef float v2f __attribute__((ext_vector_type(2)));
typedef float v8f __attribute__((ext_vector_type(8)));

constexpr int kB = 16, kN = 2048, kW = 128, kR = 4;
constexpr float kEPS = 1e-6f;

// workspace layout (float offsets)
constexpr size_t OFF_USAGE  = 0;                         // B*N
constexpr size_t OFF_ALLOC  = OFF_USAGE  + (size_t)kB*kN;      // B*N
constexpr size_t OFF_WW     = OFF_ALLOC  + (size_t)kB*kN;      // B*N
constexpr size_t OFF_SPR    = OFF_WW     + (size_t)kB*kN;      // B*R
constexpr size_t OFF_SWR    = OFF_SPR    + (size_t)kB*kR;      // B*R
constexpr size_t OFF_NK     = OFF_SWR    + (size_t)kB*kR;      // B*R
constexpr size_t OFF_PI     = OFF_NK     + (size_t)kB*kR;      // B*3*R
constexpr size_t OFF_LOGITS = OFF_PI     + (size_t)kB*3*kR;    // B*N*R
constexpr size_t OFF_LOOKUP = OFF_LOGITS + (size_t)kB*kN*kR;   // B*N*R
constexpr size_t OFF_RWTS   = OFF_LOOKUP + (size_t)kB*kN*kR;   // B*N*R
constexpr size_t OFF_MNEW   = OFF_RWTS   + (size_t)kB*kN*kR;   // B*N*W

// ---------------------------------------------------------------- K0: usage, S_pr, key norms, pi
__global__ void dnc_prep(const float* __restrict__ usage_in,
                         const float* __restrict__ prec,
                         const float* __restrict__ wwp,
                         const float* __restrict__ rwp,
                         const float* __restrict__ fg,
                         const float* __restrict__ rkeys,
                         const float* __restrict__ rmodes,
                         float* __restrict__ ws) {
  const int b = blockIdx.x, t = threadIdx.x;
  __shared__ float red[256];
  float* usage = ws + OFF_USAGE + (size_t)b * kN;
  float sp[kR] = {0.f, 0.f, 0.f, 0.f};
  for (int n = t; n < kN; n += 256) {
    float ret = 1.f, rw4[kR];
#pragma unroll
    for (int r = 0; r < kR; ++r) {
      rw4[r] = rwp[((size_t)b * kN + n) * kR + r];
      ret *= (2.f - rw4[r] * fg[b * kR + r]);
    }
    const float u0 = usage_in[b * kN + n];
    const float wp = wwp[b * kN + n];
    usage[n] = (u0 + wp - u0 * wp) * ret;
    const float pn = prec[b * kN + n];
#pragma unroll
    for (int r = 0; r < kR; ++r) sp[r] += pn * rw4[r];
  }
  for (int r = 0; r < kR; ++r) {
    red[t] = sp[r];
    __syncthreads();
    for (int s = 128; s > 0; s >>= 1) { if (t < s) red[t] += red[t + s]; __syncthreads(); }
    if (t == 0) ws[OFF_SPR + b * kR + r] = red[0];
    __syncthreads();
  }
  if (t < kR) {
    float s = 0.f;
    for (int w = 0; w < kW; ++w) {
      const float k = rkeys[((size_t)b * kW + w) * kR + t];
      s += k * k;
    }
    ws[OFF_NK + b * kR + t] = sqrtf(s);
    const float m0 = rmodes[((size_t)b * 3 + 0) * kR + t];
    const float m1 = rmodes[((size_t)b * 3 + 1) * kR + t];
    const float m2 = rmodes[((size_t)b * 3 + 2) * kR + t];
    const float mx = fmaxf(m0, fmaxf(m1, m2));
    const float e0 = __expf(m0 - mx), e1 = __expf(m1 - mx), e2 = __expf(m2 - mx);
    const float es = e0 + e1 + e2;
    ws[OFF_PI + ((size_t)b * 3 + 0) * kR + t] = e0 / es;
    ws[OFF_PI + ((size_t)b * 3 + 1) * kR + t] = e1 / es;
    ws[OFF_PI + ((size_t)b * 3 + 2) * kR + t] = e2 / es;
  }
}

// ---------------------------------------------------------------- K1: bitonic argsort + cumprod -> allocation weights
__global__ __launch_bounds__(1024) void dnc_alloc(float* __restrict__ ws) {
  const int b = blockIdx.x, t = threadIdx.x;
  __shared__ float sv[kN];
  __shared__ int   si[kN];
  __shared__ float sc[kN];
  const float* usage = ws + OFF_USAGE + (size_t)b * kN;
  for (int i = t; i < kN; i += 1024) { sv[i] = usage[i]; si[i] = i; }
  __syncthreads();
  // stable-argsort equivalent: sort (value, index) lexicographically ascending
  for (int k = 2; k <= kN; k <<= 1) {
    for (int j = k >> 1; j > 0; j >>= 1) {
      for (int i = t; i < kN; i += 1024) {
        const int l = i ^ j;
        if (l > i) {
          const float av = sv[i], bv = sv[l];
          const int ai = si[i], bi = si[l];
          const bool agtb = (av > bv) || (av == bv && ai > bi);
          const bool up = ((i & k) == 0);
          if (up ? agtb : !agtb) {
            sv[i] = bv; sv[l] = av;
            si[i] = bi; si[l] = ai;
          }
        }
      }
      __syncthreads();
    }
  }
  for (int i = t; i < kN; i += 1024) sc[i] = sv[i];
  __syncthreads();
  // inclusive product scan (Hillis-Steele), 2 elems per thread
  const int i0 = t, i1 = t + 1024;
  for (int off = 1; off < kN; off <<= 1) {
    const float t0 = (i0 >= off) ? sc[i0 - off] : 1.f;
    const float t1 = (i1 >= off) ? sc[i1 - off] : 1.f;
    __syncthreads();
    sc[i0] *= t0; sc[i1] *= t1;
    __syncthreads();
  }
  float* alloc = ws + OFF_ALLOC + (size_t)b * kN;
  for (int i = t; i < kN; i += 1024) {
    const float ex = (i == 0) ? 1.f : sc[i - 1];
    alloc[si[i]] = (1.f - sv[i]) * ex;   // scatter == take_along_axis by inverse perm
  }
}

// ---------------------------------------------------------------- K2: write content lookup + write_weight + S_wr
__global__ void dnc_write_weight(const float* __restrict__ M,
                                 const float* __restrict__ wkey,
                                 const float* __restrict__ wstr,
                                 const float* __restrict__ rwp,
                                 const float* __restrict__ ag,
                                 const float* __restrict__ wg,
                                 float* __restrict__ ws) {
  const int b = blockIdx.x, t = threadIdx.x;
  __shared__ float key[kW];
  __shared__ float red[256];
  if (t < kW) key[t] = wkey[b * kW + t];
  __syncthreads();
  red[t] = (t < kW) ? key[t] * key[t] : 0.f;
  __syncthreads();
  for (int s = 128; s > 0; s >>= 1) { if (t < s) red[t] += red[t + s]; __syncthreads(); }
  const float nK = sqrtf(red[0]);
  __syncthreads();
  const float strength = wstr[b];
  float logit[8];
#pragma unroll
  for (int q = 0; q < 8; ++q) {
    const int n = t + 256 * q;
    const float* row = M + ((size_t)b * kN + n) * kW;
    float dot = 0.f, nr = 0.f;
    for (int w = 0; w < kW; ++w) { const float x = row[w]; dot += x * key[w]; nr += x * x; }
    logit[q] = strength * dot / (sqrtf(nr) * nK + kEPS);
  }
  float lm = -3.4e38f;
#pragma unroll
  for (int q = 0; q < 8; ++q) lm = fmaxf(lm, logit[q]);
  red[t] = lm; __syncthreads();
  for (int s = 128; s > 0; s >>= 1) { if (t < s) red[t] = fmaxf(red[t], red[t + s]); __syncthreads(); }
  const float gmax = red[0]; __syncthreads();
  float ls = 0.f;
#pragma unroll
  for (int q = 0; q < 8; ++q) ls += __expf(logit[q] - gmax);
  red[t] = ls; __syncthreads();
  for (int s = 128; s > 0; s >>= 1) { if (t < s) red[t] += red[t + s]; __syncthreads(); }
  const float gsum = red[0]; __syncthreads();
  const float a = ag[b], g = wg[b];
  const float* alloc = ws + OFF_ALLOC + (size_t)b * kN;
  float* wwv = ws + OFF_WW + (size_t)b * kN;
  float swr[kR] = {0.f, 0.f, 0.f, 0.f};
#pragma unroll
  for (int q = 0; q < 8; ++q) {
    const int n = t + 256 * q;
    const float lw = __expf(logit[q] - gmax) / gsum;
    const float w_ = g * (a * alloc[n] + (1.f - a) * lw);
    wwv[n] = w_;
#pragma unroll
    for (int r = 0; r < kR; ++r) swr[r] += w_ * rwp[((size_t)b * kN + n) * kR + r];
  }
  for (int r = 0; r < kR; ++r) {
    red[t] = swr[r]; __syncthreads();
    for (int s = 128; s > 0; s >>= 1) { if (t < s) red[t] += red[t + s]; __syncthreads(); }
    if (t == 0) ws[OFF_SWR + b * kR + r] = red[0];
    __syncthreads();
  }
}

// ---------------------------------------------------------------- K3: memory_new + read-key cosine logits
__global__ void dnc_mem(const float* __restrict__ M,
                        const float* __restrict__ erase,
                        const float* __restrict__ wvec,
                        const float* __restrict__ rkeys,
                        const float* __restrict__ rstr,
                        float* __restrict__ ws) {
  const int blk = blockIdx.x;
  const int b = blk / (kN / 8);
  const int nbase = (blk % (kN / 8)) * 8;
  const int t = threadIdx.x, wave = t >> 5, lane = t & 31;
  __shared__ float key[kW * kR];
  __shared__ float sK[kR], snK[kR];
  if (t < kW * kR) key[t] = rkeys[(size_t)b * kW * kR + t];
  if (t < kR) { sK[t] = rstr[b * kR + t]; snK[t] = ws[OFF_NK + b * kR + t]; }
  __syncthreads();
  const int n = nbase + wave;
  const float* row = M + ((size_t)b * kN + n) * kW;
  float* nrow = ws + OFF_MNEW + ((size_t)b * kN + n) * kW;
  const float wwn = ws[OFF_WW + (size_t)b * kN + n];
  float d0 = 0.f, d1 = 0.f, d2 = 0.f, d3 = 0.f, nr = 0.f;
#pragma unroll
  for (int q = 0; q < 4; ++q) {
    const int w = lane + 32 * q;
    const float x = row[w] * (1.f - wwn * erase[b * kW + w]) + wwn * wvec[b * kW + w];
    nrow[w] = x;
    nr += x * x;
    d0 += x * key[w * kR + 0];
    d1 += x * key[w * kR + 1];
    d2 += x * key[w * kR + 2];
    d3 += x * key[w * kR + 3];
  }
  for (int m = 16; m > 0; m >>= 1) {
    nr += __shfl_xor(nr, m, 32);
    d0 += __shfl_xor(d0, m, 32);
    d1 += __shfl_xor(d1, m, 32);
    d2 += __shfl_xor(d2, m, 32);
    d3 += __shfl_xor(d3, m, 32);
  }
  if (lane == 0) {
    const float nm = sqrtf(nr);
    float* lg = ws + OFF_LOGITS + ((size_t)b * kN + n) * kR;
    lg[0] = sK[0] * d0 / (nm * snK[0] + kEPS);
    lg[1] = sK[1] * d1 / (nm * snK[1] + kEPS);
    lg[2] = sK[2] * d2 / (nm * snK[2] + kEPS);
    lg[3] = sK[3] * d3 / (nm * snK[3] + kEPS);
  }
}

// ---------------------------------------------------------------- K4: softmax over N per (b, r)
__global__ void dnc_softmax_r(float* __restrict__ ws) {
  const int b = blockIdx.x >> 2, r = blockIdx.x & 3, t = threadIdx.x;
  __shared__ float red[256];
  const float* lg = ws + OFF_LOGITS + (size_t)b * kN * kR + r;
  float lm = -3.4e38f;
  for (int n = t; n < kN; n += 256) lm = fmaxf(lm, lg[(size_t)n * kR]);
  red[t] = lm; __syncthreads();
  for (int s = 128; s > 0; s >>= 1) { if (t < s) red[t] = fmaxf(red[t], red[t + s]); __syncthreads(); }
  const float gmax = red[0]; __syncthreads();
  float ls = 0.f;
  for (int n = t; n < kN; n += 256) ls += __expf(lg[(size_t)n * kR] - gmax);
  red[t] = ls; __syncthreads();
  for (int s = 128; s > 0; s >>= 1) { if (t < s) red[t] += red[t + s]; __syncthreads(); }
  const float gsum = red[0]; __syncthreads();
  float* lk = ws + OFF_LOOKUP + (size_t)b * kN * kR + r;
  for (int n = t; n < kN; n += 256) lk[(size_t)n * kR] = __expf(lg[(size_t)n * kR] - gmax) / gsum;
}

// ---------------------------------------------------------------- K5: WMMA link pass -> read_weights
// fwd = link_new @ rw ; bwd = link_new^T @ rw, fused via L @ [rw | ww*rw] and L^T @ [rw | ww*rw].
// Branchless B-fragment (EXEC stays all-1s) + two interleaved K-chains to hide WMMA latency.
__global__ void dnc_link(const float* __restrict__ L,
                         const float* __restrict__ rwp,
                         const float* __restrict__ prec,
                         float* __restrict__ ws) {
  const int blk = blockIdx.x;
  const int b = blk / (kN / 128);
  const int tile = blk % (kN / 128);
  const int t = threadIdx.x, wave = t >> 5, lane = t & 31;
  __shared__ float lds_rw[kN * kR];  // 32 KB
  __shared__ float lds_ww[kN];       //  8 KB
  __shared__ float lds_p[kN];        //  8 KB
  const float* wwv = ws + OFF_WW + (size_t)b * kN;
  for (int i = t; i < kN * kR; i += 256) lds_rw[i] = rwp[(size_t)b * kN * kR + i];
  for (int i = t; i < kN; i += 256) { lds_ww[i] = wwv[i]; lds_p[i] = prec[(size_t)b * kN + i]; }
  __syncthreads();

  const float* Lb = L + (size_t)b * kN * kN;
  const int n0 = tile * 128 + wave * 16;
  const int rl = lane & 15;              // A/B row-within-halfwave / D column
  const int kb = (lane < 16) ? 0 : 2;    // K sub-offset held by this half-wave
  const int j3 = rl & 3;                 // read-head column this lane sources
  // loop-invariant column selectors: cols 0..3 -> rw, cols 4..7 -> ww*rw, cols 8..15 -> 0
  const float c1 = (rl < 4) ? 1.f : 0.f;
  const float c2 = (rl >= 4 && rl < 8) ? 1.f : 0.f;
  const float* rowF = Lb + (size_t)(n0 + rl) * kN;
  const float* colB = Lb + (n0 + rl);

  v8f accF0 = {0.f,0.f,0.f,0.f,0.f,0.f,0.f,0.f};
  v8f accB0 = {0.f,0.f,0.f,0.f,0.f,0.f,0.f,0.f};
  v8f accF1 = {0.f,0.f,0.f,0.f,0.f,0.f,0.f,0.f};
  v8f accB1 = {0.f,0.f,0.f,0.f,0.f,0.f,0.f,0.f};

  for (int m0 = 0; m0 < kN; m0 += 8) {
    const int mk0 = m0 + kb;       // chain 0: K-chunk [m0, m0+4)
    const int mk1 = m0 + 4 + kb;   // chain 1: K-chunk [m0+4, m0+8)

    // A (fwd): contiguous pairs from this lane's row
    const v2f aF0 = *reinterpret_cast<const v2f*>(rowF + mk0);
    const v2f aF1 = *reinterpret_cast<const v2f*>(rowF + mk1);
    // A (bwd): transposed slab — 16 consecutive lanes hit 64B segments of rows mk..mk+1
    v2f aB0, aB1;
    aB0.x = colB[(size_t)mk0 * kN];
    aB0.y = colB[(size_t)(mk0 + 1) * kN];
    aB1.x = colB[(size_t)mk1 * kN];
    aB1.y = colB[(size_t)(mk1 + 1) * kN];

    // B fragments, branchless: scale = c1 + c2*ww[m]  (0 for cols 8..15)
    const v2f wwp0 = *reinterpret_cast<const v2f*>(lds_ww + mk0);  // ww[mk0], ww[mk0+1]
    const v2f wwp1 = *reinterpret_cast<const v2f*>(lds_ww + mk1);
    v2f bf0, bf1;
    bf0.x = lds_rw[mk0 * kR + j3] * (c1 + c2 * wwp0.x);
    bf0.y = lds_rw[(mk0 + 1) * kR + j3] * (c1 + c2 * wwp0.y);
    bf1.x = lds_rw[mk1 * kR + j3] * (c1 + c2 * wwp1.x);
    bf1.y = lds_rw[(mk1 + 1) * kR + j3] * (c1 + c2 * wwp1.y);

    __builtin_prefetch(rowF + mk0 + 64, 0, 3);   // global_prefetch_b8, keep near

    accF0 = __builtin_amdgcn_wmma_f32_16x16x4_f32(false, aF0, false, bf0, (short)0, accF0, false, false);
    accB0 = __builtin_amdgcn_wmma_f32_16x16x4_f32(false, aB0, false, bf0, (short)0, accB0, false, false);
    accF1 = __builtin_amdgcn_wmma_f32_16x16x4_f32(false, aF1, false, bf1, (short)0, accF1, false, false);
    accB1 = __builtin_amdgcn_wmma_f32_16x16x4_f32(false, aB1, false, bf1, (short)0, accB1, false, false);
  }
  const v8f accF = accF0 + accF1;
  const v8f accB = accB0 + accB1;

  const float* Spr = ws + OFF_SPR + (size_t)b * kR;
  const float* Swr = ws + OFF_SWR + (size_t)b * kR;
  const float* pi = ws + OFF_PI + (size_t)b * 3 * kR;
#pragma unroll
  for (int v = 0; v < 8; ++v) {
    const float F1 = accF[v];
    const float F2 = __shfl_xor(F1, 4, 32);   // column r+4 lives in lane^4
    const float T1 = accB[v];
    const float T2 = __shfl_xor(T1, 4, 32);
    if (rl < 4) {
      const int r = rl;
      const int nl = v + ((lane < 16) ? 0 : 8);
      const int n = n0 + nl;
      const float wwn = lds_ww[n], pn = lds_p[n];
      const float Lnn = Lb[(size_t)n * kN + n];
      const float rwnr = lds_rw[n * kR + r];
      const float dia = ((1.f - 2.f * wwn) * Lnn + wwn * pn) * rwnr;  // remove m==n term
      const float fwd = (1.f - wwn) * F1 - F2 + wwn * Spr[r] - dia;
      const float bwd = (1.f - wwn) * T1 - T2 + pn * Swr[r] - dia;
      const float lr = ws[OFF_LOOKUP + ((size_t)b * kN + n) * kR + r];
      const float p0 = pi[0 * kR + r], p1 = pi[1 * kR + r], p2 = pi[2 * kR + r];
      ws[OFF_RWTS + ((size_t)b * kN + n) * kR + r] = bwd * p0 + lr * p1 + fwd * p2;
    }
  }
}

// ---------------------------------------------------------------- K6: read_vectors = Mnew^T @ read_weights
__global__ __launch_bounds__(512) void dnc_read(const float* __restrict__ ws,
                                                float* __restrict__ out) {
  const int b = blockIdx.x, t = threadIdx.x;
  const int w = t >> 2, r = t & 3;
  __shared__ float rch[256 * kR];
  float acc = 0.f;
  for (int nb = 0; nb < kN; nb += 256) {
    for (int e = t; e < 256 * kR; e += 512)
      rch[e] = ws[OFF_RWTS + ((size_t)b * kN + nb) * kR + e];
    __syncthreads();
    const float* Mn = ws + OFF_MNEW + ((size_t)b * kN + nb) * kW;
    for (int i = 0; i < 256; ++i) acc += Mn[(size_t)i * kW + w] * rch[i * kR + r];
    __syncthreads();
  }
  out[((size_t)b * kW + w) * kR + r] = acc;
}

// ----------------------------------------------------------------
extern "C" void kernel_launch(void* const* d_in, const int* in_sizes, int n_in,
                              void* d_out, int out_size, void* d_ws, size_t ws_size,
                              hipStream_t stream) {
  const float* M     = (const float*)d_in[0];   // (B,N,W)
  const float* usage = (const float*)d_in[1];   // (B,N)
  const float* prec  = (const float*)d_in[2];   // (B,N)
  const float* L     = (const float*)d_in[3];   // (B,N,N)
  const float* wwp   = (const float*)d_in[4];   // (B,N)
  const float* rwp   = (const float*)d_in[5];   // (B,N,R)
  const float* rkeys = (const float*)d_in[6];   // (B,W,R)
  const float* rstr  = (const float*)d_in[7];   // (B,R)
  const float* wkey  = (const float*)d_in[8];   // (B,W,1)
  const float* wstr  = (const float*)d_in[9];   // (B,1)
  const float* fg    = (const float*)d_in[10];  // (B,R)
  const float* ag    = (const float*)d_in[11];  // (B,1)
  const float* wg    = (const float*)d_in[12];  // (B,1)
  const float* wvec  = (const float*)d_in[13];  // (B,W)
  const float* erase = (const float*)d_in[14];  // (B,W)
  const float* rmod  = (const float*)d_in[15];  // (B,3,R)
  float* ws = (float*)d_ws;
  float* out = (float*)d_out;

  dnc_prep<<<kB, 256, 0, stream>>>(usage, prec, wwp, rwp, fg, rkeys, rmod, ws);
  dnc_alloc<<<kB, 1024, 0, stream>>>(ws);
  dnc_write_weight<<<kB, 256, 0, stream>>>(M, wkey, wstr, rwp, ag, wg, ws);
  dnc_mem<<<kB * (kN / 8), 256, 0, stream>>>(M, erase, wvec, rkeys, rstr, ws);
  dnc_softmax_r<<<kB * kR, 256, 0, stream>>>(ws);
  dnc_link<<<kB * (kN / 128), 256, 0, stream>>>(L, rwp, prec, ws);
  dnc_read<<<kB, 512, 0, stream>>>(ws, out);
}